// Gdatt_56942676410885
// MI455X (gfx1250) — compile-verified
//
#include <hip/hip_runtime.h>
#include <hip/hip_bf16.h>

// out = expm(wsum * logm(V)) with wsum ≡ 1  =>  out = V = Wv^T X Wv.
// Batched congruence transform, fp32 WMMA (V_WMMA_F32_16X16X4_F32),
// X/Wv staged into LDS via CDNA5 async global->LDS copies (ASYNCcnt).

typedef __attribute__((ext_vector_type(2))) float v2f;
typedef __attribute__((ext_vector_type(8))) float v8f;
typedef __attribute__((address_space(3))) float* lds_fp;

#define DI 128   // D_IN
#define DO 64    // D_OUT

// LDS row strides (floats), padded so fragment gathers are bank-conflict-free:
// need (2*stride mod 64) in [16,48] for the half-wave row split (K and K+2).
#define XS 136   // X rows (128 cols)
#define WS 72    // Wv rows (64 cols)
#define TS 132   // T rows (128 cols); stride mod 64 == 4 -> column gathers conflict-free

// Async 16B global->LDS copy, no VGPR data round-trip (tracked by ASYNCcnt).
__device__ __forceinline__ void async_cp16(lds_fp dst, const float* src) {
  asm volatile("global_load_async_to_lds_b128 %0, %1, off"
               :: "v"(dst), "v"(src) : "memory");
}

__launch_bounds__(128, 1)
__global__ void spd_attn_congruence(const float* __restrict__ x,
                                    const float* __restrict__ wv,
                                    float* __restrict__ out) {
  __shared__ __align__(16) float Xs[DI * XS];   // 69,632 B
  __shared__ __align__(16) float Ws[DI * WS];   // 36,864 B
  __shared__ __align__(16) float Ts[DO * TS];   // 33,792 B

  const int tid  = threadIdx.x;
  const int lane = tid & 31;
  const int wave = tid >> 5;          // 4 waves, wave w owns output rows 16w..16w+15
  const int lm   = lane & 15;
  const int half = lane >> 4;         // A-frag: lanes 0-15 hold K,K+1; lanes 16-31 hold K+2,K+3

  const float* xg = x + (size_t)blockIdx.x * (DI * DI);

  // --- stage X (128x128) and Wv (128x64) into LDS: async b128, no VGPR traffic ---
  for (int i = tid; i < (DI * DI / 4); i += 128) {
    const int r = i >> 5, c4 = (i & 31) << 2;
    async_cp16((lds_fp)&Xs[r * XS + c4], xg + r * DI + c4);
  }
  for (int i = tid; i < (DI * DO / 4); i += 128) {
    const int r = i >> 4, c4 = (i & 15) << 2;
    async_cp16((lds_fp)&Ws[r * WS + c4], wv + r * DO + c4);
  }
  asm volatile("s_wait_asynccnt 0x0" ::: "memory");  // my async writes landed in LDS
  __syncthreads();                                   // everyone's writes visible

  const int m0 = wave << 4;

  // --- Stage 1: T[m0..m0+15, :] = (Wv^T) * X ---
  // A[m,k] = Wv[k, m] (transposed read from LDS), hoisted for all 32 K-steps.
  v2f af[32];
#pragma unroll
  for (int ks = 0; ks < 32; ++ks) {
    const int kb = (ks << 2) + (half << 1);
    af[ks].x = Ws[kb * WS + m0 + lm];
    af[ks].y = Ws[(kb + 1) * WS + m0 + lm];
  }

  for (int nt = 0; nt < 8; ++nt) {
    const int n0 = nt << 4;
    v8f acc = {0.f, 0.f, 0.f, 0.f, 0.f, 0.f, 0.f, 0.f};
#pragma unroll
    for (int ks = 0; ks < 32; ++ks) {
      const int kb = (ks << 2) + (half << 1);
      v2f b;
      b.x = Xs[kb * XS + n0 + lm];          // B[k,n] = X[k, n0+n]
      b.y = Xs[(kb + 1) * XS + n0 + lm];
      acc = __builtin_amdgcn_wmma_f32_16x16x4_f32(
          /*neg_a=*/false, af[ks], /*neg_b=*/false, b,
          /*c_mod=*/(short)0, acc, /*reuse_a=*/false, /*reuse_b=*/false);
    }
    // C/D layout: VGPR r -> (M = r + 8*half, N = lm)
#pragma unroll
    for (int r = 0; r < 8; ++r)
      Ts[(m0 + r + 8 * half) * TS + n0 + lm] = acc[r];
  }

  // --- Stage 2: Out[m0..m0+15, :] = T_strip * Wv ---
  // Wave reads only the strip it wrote itself -> no inter-wave barrier needed.
  v2f tf[32];
#pragma unroll
  for (int ks = 0; ks < 32; ++ks) {
    const int kb = (ks << 2) + (half << 1);
    tf[ks].x = Ts[(m0 + lm) * TS + kb];
    tf[ks].y = Ts[(m0 + lm) * TS + kb + 1];
  }

  float* og = out + (size_t)blockIdx.x * (DO * DO);
  for (int nt = 0; nt < 4; ++nt) {
    const int n0 = nt << 4;
    v8f acc = {0.f, 0.f, 0.f, 0.f, 0.f, 0.f, 0.f, 0.f};
#pragma unroll
    for (int ks = 0; ks < 32; ++ks) {
      const int kb = (ks << 2) + (half << 1);
      v2f b;
      b.x = Ws[kb * WS + n0 + lm];          // B[k,n] = Wv[k, n0+n] (non-transposed)
      b.y = Ws[(kb + 1) * WS + n0 + lm];
      acc = __builtin_amdgcn_wmma_f32_16x16x4_f32(
          false, tf[ks], false, b, (short)0, acc, false, false);
    }
#pragma unroll
    for (int r = 0; r < 8; ++r)
      og[(m0 + r + 8 * half) * DO + n0 + lm] = acc[r];
  }
}

extern "C" void kernel_launch(void* const* d_in, const int* in_sizes, int n_in,
                              void* d_out, int out_size, void* d_ws, size_t ws_size,
                              hipStream_t stream) {
  (void)n_in; (void)out_size; (void)d_ws; (void)ws_size;
  const float* x  = (const float*)d_in[0];
  // d_in[1] = Wq, d_in[2] = Wk: mathematically dead (wsum == 1 collapse).
  const float* wv = (const float*)d_in[3];
  float* out = (float*)d_out;

  const int nmat = in_sizes[0] / (DI * DI);   // BS*M = 2048
  spd_attn_congruence<<<dim3(nmat), dim3(128), 0, stream>>>(x, wv, out);
}